// DistanceSampling_4483945857080
// MI455X (gfx1250) — compile-verified
//
#include <hip/hip_runtime.h>
#include <stdint.h>

// Problem constants (fixed by the reference): x = (1, 256, 512, 512) fp32,
// 2x2 stride-2 tiling -> 256x256 patches, out = (1, 256, 65536) fp32.
namespace {
constexpr int Cc   = 256;
constexpr int Hh   = 512;
constexpr int Ww   = 512;
constexpr int FH   = 256;            // patch rows
constexpr int FW   = 256;            // patch cols
constexpr int HW   = Hh * Ww;        // per-channel plane elements
constexpr int Ltot = FH * FW;        // total patches
constexpr float EPSF = 1e-6f;
constexpr int TJ     = 32;           // patches per block (one per lane, wave32)
constexpr int CHUNKS = FW / TJ;      // 8 column chunks per patch row
constexpr int NBUF   = 16;           // async ring depth: 15 loads in flight/wave
constexpr int BUFB   = 4 * TJ * 4;   // bytes per ring slot (512 B)
}

// CDNA5 async global->LDS copy, 16B per lane, tracked by ASYNCcnt.
__device__ __forceinline__ void async_b128_to_lds(uint32_t lds_byte, const float* g) {
  asm volatile("global_load_async_to_lds_b128 %0, %1, off"
               :: "v"(lds_byte), "v"(g)
               : "memory");
}

__global__ __launch_bounds__(TJ) void patchdist_kernel(const float* __restrict__ x,
                                                       float* __restrict__ out) {
  // 16-slot ring of LDS tiles: per slot 128 floats = 64 (row0) + 64 (row1). 8 KB total.
  __shared__ float buf[NBUF][4 * TJ];

  const int tid = threadIdx.x;                 // patch within chunk / copy lane
  const int i   = blockIdx.x / CHUNKS;         // patch row
  const int j0  = (blockIdx.x % CHUNKS) * TJ;  // first patch col of this chunk
  const int y0  = 2 * i;

  // Global base of this chunk's 2-row window for channel 0.
  const float* base = x + (size_t)y0 * Ww + 2 * j0;
  // Lane -> source offset (floats): lanes 0..15 cover row y0 (64 floats),
  // lanes 16..31 cover row y0+1 (64 floats). 16B per lane, coalesced.
  const size_t goff = (tid < 16) ? (size_t)(tid * 4)
                                 : ((size_t)Ww + (size_t)(tid - 16) * 4);

  const uint32_t lbase = (uint32_t)(uintptr_t)&buf[0][0];
  const uint32_t loff  = (uint32_t)tid * 16u;

  // Prime the pipeline: channels 0..14 into slots 0..14.
#pragma unroll
  for (int c = 0; c < NBUF - 1; ++c) {
    async_b128_to_lds(lbase + (uint32_t)c * BUFB + loff,
                      base + (size_t)c * HW + goff);
  }

  float s0 = 0.f, s1 = 0.f, s2 = 0.f, s3 = 0.f;

  for (int c = 0; c < Cc; ++c) {
    const int pf = c + NBUF - 1;
    if (pf < Cc) {
      // Issue prefetch for channel c+15 into ring slot (c-1)%16; that slot's
      // values were consumed (FMA-forced DS waits) at iteration c-1, so the
      // write-after-read is safe within this single wave.
      async_b128_to_lds(lbase + (uint32_t)(pf & (NBUF - 1)) * BUFB + loff,
                        base + (size_t)pf * HW + goff);
      asm volatile("s_wait_asynccnt 15" ::: "memory");  // oldest (channel c) done
    } else {
      asm volatile("s_wait_asynccnt 0" ::: "memory");   // tail drain
    }
    // Single-wave workgroup: ASYNCcnt wait alone makes the tile visible; no barrier.
    const float* b = buf[c & (NBUF - 1)];
    const float a0 = b[2 * tid];              // (kh=0, kw=0)
    const float a1 = b[2 * tid + 1];          // (kh=0, kw=1)
    const float a2 = b[2 * TJ + 2 * tid];     // (kh=1, kw=0)
    const float a3 = b[2 * TJ + 2 * tid + 1]; // (kh=1, kw=1)
    const float m  = (a0 + a1 + a2 + a3) * 0.25f;
    const float d0 = a0 - m + EPSF;
    const float d1 = a1 - m + EPSF;
    const float d2 = a2 - m + EPSF;
    const float d3 = a3 - m + EPSF;
    s0 = fmaf(d0, d0, s0);
    s1 = fmaf(d1, d1, s1);
    s2 = fmaf(d2, d2, s2);
    s3 = fmaf(d3, d3, s3);
  }

  // First-occurrence argmax over k (strict > keeps the lowest index on ties);
  // sqrt is monotone so comparing squared sums is exact.
  int idx = 0;
  float best = s0;
  if (s1 > best) { best = s1; idx = 1; }
  if (s2 > best) { best = s2; idx = 2; }
  if (s3 > best) { best = s3; idx = 3; }

  // Phase 2: gather the winning element per channel (re-reads hit the 192 MB L2),
  // coalesced writes to out[c*L + l]. Deep unroll keeps ~32 loads in flight.
  const int l = i * FW + (j0 + tid);
  const float* g = x + (size_t)y0 * Ww + 2 * (j0 + tid);
#pragma unroll 16
  for (int c = 0; c < Cc; ++c) {
    const float* gp = g + (size_t)c * HW;
    const float a = gp[0];
    const float bb = gp[1];
    const float p = gp[Ww];
    const float q = gp[Ww + 1];
    const float sel = (idx == 0) ? a : (idx == 1) ? bb : (idx == 2) ? p : q;
    out[(size_t)c * Ltot + l] = sel;
  }
}

extern "C" void kernel_launch(void* const* d_in, const int* in_sizes, int n_in,
                              void* d_out, int out_size, void* d_ws, size_t ws_size,
                              hipStream_t stream) {
  (void)in_sizes; (void)n_in; (void)out_size; (void)d_ws; (void)ws_size;
  const float* x = (const float*)d_in[0];
  float* out = (float*)d_out;
  patchdist_kernel<<<dim3(FH * CHUNKS), dim3(TJ), 0, stream>>>(x, out);
}